// DynamicAssign_65498251264574
// MI455X (gfx1250) — compile-verified
//
#include <hip/hip_runtime.h>
#include <hip/hip_bf16.h>
#include <stdint.h>

// Problem shape (fixed by reference setup_inputs)
#define WB 16
#define WP 2048
#define WG 32
#define WD 78
#define WN 72
#define LHALF 15.0f

typedef uint32_t u32x4 __attribute__((ext_vector_type(4)));
typedef int      i32x8 __attribute__((ext_vector_type(8)));
typedef int      i32x4 __attribute__((ext_vector_type(4)));

#if defined(__has_builtin)
#  if __has_builtin(__builtin_amdgcn_tensor_load_to_lds)
#    define HAVE_TDM 1
#  endif
#endif
#ifndef HAVE_TDM
#  define HAVE_TDM 0
#endif

// Monotonic float -> uint32 mapping (total order matching float <)
__device__ __forceinline__ unsigned ordf(float f) {
  unsigned u = __float_as_uint(f);
  return (u & 0x80000000u) ? ~u : (u | 0x80000000u);
}

// ---------------------------------------------------------------------------
// Kernel 1: pairwise cost + masked line IoU.
//   grid (P/256, B), block 256. Targets[b] (32x78 f32 = 9984B) staged into LDS
//   via the Tensor Data Mover (one descriptor per workgroup, issued by wave 0).
//   Pred line points kept in registers and reused across all 32 GTs.
// ---------------------------------------------------------------------------
__global__ __launch_bounds__(256) void k_cost(
    const float* __restrict__ preds, const float* __restrict__ targets,
    const unsigned char* __restrict__ masks, const int* __restrict__ img_w_p,
    float* __restrict__ cost, float* __restrict__ iou_out) {
  const int b = blockIdx.y;
  const int p = blockIdx.x * 256 + threadIdx.x;
  const int img_w = *img_w_p;
  const float scale = (float)(img_w - 1);
  const float img_wf = (float)img_w;

  __shared__ float T[WG * WD];
  const float* tgt_base = targets + (size_t)b * WG * WD;

#if HAVE_TDM
  if (threadIdx.x < 32) {  // one wave issues the TDM op (TDM ignores EXEC)
    uint32_t lds_addr = (uint32_t)(uintptr_t)(&T[0]);  // low 32b of generic addr = LDS byte offset
    unsigned long long ga = (unsigned long long)(uintptr_t)tgt_base;
    u32x4 g0;
    g0[0] = 1u;                                   // count=1, user mode
    g0[1] = lds_addr;                             // lds_addr (bytes)
    g0[2] = (uint32_t)(ga & 0xFFFFFFFFull);       // global_addr[31:0]
    g0[3] = (uint32_t)((ga >> 32) & 0x1FFFFFFull) // global_addr[56:32]
            | (2u << 30);                         // type=2 ("image")
    i32x8 g1;
    g1[0] = (int)(2u << 16);                  // wg_mask=0, data_size=2 (4B)
    g1[1] = (int)((uint32_t)WD << 16);        // tensor_dim0[15:0] = 78
    g1[2] = (int)((uint32_t)WG << 16);        // tensor_dim0 hi=0 | tensor_dim1[15:0]=32
    g1[3] = (int)((uint32_t)WD << 16);        // tensor_dim1 hi=0 | tile_dim0=78
    g1[4] = (int)WG;                          // tile_dim1=32, tile_dim2=0
    g1[5] = (int)WD;                          // tensor_dim0_stride lo = 78
    g1[6] = (int)((uint32_t)WD << 16);        // stride0 hi=0 | tensor_dim1_stride lo16 = 78
    g1[7] = 0;                                // tensor_dim1_stride hi
    i32x4 z4 = {0, 0, 0, 0};                  // groups 2/3 unused (2D tensor)
    i32x8 z8 = {0, 0, 0, 0, 0, 0, 0, 0};      // trailing group (6-arg toolchain form)
    __builtin_amdgcn_tensor_load_to_lds(g0, g1, z4, z4, z8, 0);
    __builtin_amdgcn_s_wait_tensorcnt(0);
  }
  __syncthreads();
#else
  for (int i = threadIdx.x; i < WG * WD; i += 256) T[i] = tgt_base[i];
  __syncthreads();
#endif

  if (p >= WP) return;  // no further barriers below
  const float* pr = preds + ((size_t)b * WP + p) * WD;
  __builtin_prefetch(pr, 0, 3);  // global_prefetch_b8

  // classification cost: -log softmax(preds[0:2])[1], clipped at 1e-8
  float x0 = pr[0], x1 = pr[1];
  float mx = fmaxf(x0, x1);
  float e0 = expf(x0 - mx), e1 = expf(x1 - mx);
  float cls_cost = -logf(fmaxf(e1 / (e0 + e1), 1e-8f));
  float pr2 = pr[2], pr3 = pr[3], pr4 = pr[4];

  float r[WN];
#pragma unroll
  for (int n = 0; n < WN; ++n) r[n] = pr[6 + n] * scale;

  for (int g = 0; g < WG; ++g) {
    const float* Tg = &T[g * WD];
    float reg = fabsf(pr3 - Tg[3]) + fabsf(pr2 - Tg[2]) + fabsf(pr4 - Tg[4]);
    float ovr = 0.f, uni = 0.f;
#pragma unroll
    for (int n = 0; n < WN; ++n) {
      float t = Tg[6 + n] * scale;
      bool inv = (t < 0.f) || (t >= img_wf);
      float d = fabsf(r[n] - t);
      ovr += inv ? 0.f : (2.f * LHALF - d);  // min(px2,tx2)-max(px1,tx1)
      uni += inv ? 0.f : (2.f * LHALF + d);  // max(px2,tx2)-min(px1,tx1)
    }
    float iou = ovr / (uni + 1e-9f);
    if (iou != iou) iou = 0.f;  // nan_to_num
    bool valid = masks[b * WG + g] != 0;
    float maskf = valid ? 1.f : 0.f;
    iou *= maskf;
    float iou_cost = valid ? -logf(fmaxf(iou, 1e-8f)) : 0.f;
    float c = 3.f * cls_cost + 3.f * reg + 3.f * iou_cost + 1e5f * (1.f - maskf);
    size_t idx = (size_t)(b * WG + g) * WP + p;
    cost[idx] = c;
    iou_out[idx] = iou;
  }
}

// ---------------------------------------------------------------------------
// Kernel 2: dynamic_k[b,g] = clip(trunc(sum of top-10 IoU over P), 1, P).
//   One 256-thread block per (b,g); 10 rounds of max-extraction in LDS.
// ---------------------------------------------------------------------------
__global__ __launch_bounds__(256) void k_dynk(const float* __restrict__ iou,
                                              int* __restrict__ ks) {
  const int g = blockIdx.x, b = blockIdx.y, t = threadIdx.x;
  __shared__ float vals[WP];
  __shared__ unsigned long long red[256];
  __shared__ float ssum;
  const float* row = iou + (size_t)(b * WG + g) * WP;
  for (int i = t; i < WP; i += 256) vals[i] = row[i];
  if (t == 0) ssum = 0.f;
  __syncthreads();
  for (int it = 0; it < 10; ++it) {
    unsigned long long best = 0ull;
    for (int i = t; i < WP; i += 256) {
      unsigned long long k = ((unsigned long long)ordf(vals[i]) << 32) | (unsigned)i;
      if (k > best) best = k;
    }
    red[t] = best;
    __syncthreads();
    for (int s = 128; s > 0; s >>= 1) {
      if (t < s) { unsigned long long o = red[t + s]; if (o > red[t]) red[t] = o; }
      __syncthreads();
    }
    if (t == 0) {
      int idx = (int)(red[0] & 0xFFFFFFFFull);
      ssum += vals[idx];
      vals[idx] = -__builtin_inff();  // remove extracted element
    }
    __syncthreads();
  }
  if (t == 0) {
    int k = (int)ssum;  // astype(int32): trunc toward zero
    if (k < 1) k = 1;
    if (k > WP) k = WP;
    ks[b * WG + g] = k;
  }
}

// ---------------------------------------------------------------------------
// Kernel 3: sequential per-GT assignment with min-cost conflict resolution.
//   One 1024-thread block per batch (each thread owns priors t and t+1024).
//   "rank < k" == "composite key among k smallest" (key = ord(cost)<<11 | p,
//   which reproduces stable argsort tie-breaking). k <= 10 always, so k
//   rounds of min-extraction give the k-th-smallest threshold cheaply.
// ---------------------------------------------------------------------------
__global__ __launch_bounds__(1024) void k_assign(
    const float* __restrict__ cost, const int* __restrict__ ks,
    const unsigned char* __restrict__ masks, float* __restrict__ out) {
  const int b = blockIdx.x, t = threadIdx.x;
  __shared__ unsigned long long keys[WP];
  __shared__ unsigned long long red[1024];
  __shared__ unsigned long long thr;
  float minc0 = 1e8f, minc1 = 1e8f;
  int m0 = -1, m1 = -1;
  const int p0 = t, p1 = t + 1024;

  for (int g = 0; g < WG; ++g) {
    const float* row = cost + (size_t)(b * WG + g) * WP;
    float c0 = row[p0], c1 = row[p1];
    unsigned long long k0 = ((unsigned long long)ordf(c0) << 11) | (unsigned)p0;
    unsigned long long k1 = ((unsigned long long)ordf(c1) << 11) | (unsigned)p1;
    keys[p0] = k0;
    keys[p1] = k1;
    int kg = ks[b * WG + g];
    if (kg > WP) kg = WP;
    bool valid = masks[b * WG + g] != 0;
    __syncthreads();

    for (int it = 0; it < kg; ++it) {  // kg <= 10 in practice (IoU <= 1)
      unsigned long long a = keys[t], c = keys[t + 1024];
      red[t] = (a < c) ? a : c;
      __syncthreads();
      for (int s = 512; s > 0; s >>= 1) {
        if (t < s) { unsigned long long o = red[t + s]; if (o < red[t]) red[t] = o; }
        __syncthreads();
      }
      if (t == 0) {
        unsigned long long M = red[0];
        if (it == kg - 1) thr = M;                       // k-th smallest key
        else keys[(unsigned)(M & (WP - 1))] = ~0ull;     // remove current min
      }
      __syncthreads();
    }
    unsigned long long th = thr;
    if (valid) {
      if (k0 <= th && c0 < minc0) { m0 = g; minc0 = c0; }
      if (k1 <= th && c1 < minc1) { m1 = g; minc1 = c1; }
    }
    __syncthreads();  // keys/red/thr reused next g
  }

  out[(size_t)b * WP + p0] = (m0 >= 0) ? 1.f : 0.f;
  out[(size_t)b * WP + p1] = (m1 >= 0) ? 1.f : 0.f;
  float* om = out + (size_t)WB * WP;
  om[(size_t)b * WP + p0] = (float)m0;
  om[(size_t)b * WP + p1] = (float)m1;
}

// ---------------------------------------------------------------------------
extern "C" void kernel_launch(void* const* d_in, const int* in_sizes, int n_in,
                              void* d_out, int out_size, void* d_ws, size_t ws_size,
                              hipStream_t stream) {
  (void)in_sizes; (void)n_in; (void)out_size; (void)ws_size;
  const float* preds = (const float*)d_in[0];           // (B,P,78) f32
  const float* targets = (const float*)d_in[1];         // (B,G,78) f32
  const unsigned char* masks = (const unsigned char*)d_in[2];  // (B,G) bool (1B)
  const int* img_w = (const int*)d_in[3];               // scalar
  // d_in[4] = img_h (unused by reference math)
  float* out = (float*)d_out;                           // [mask(B,P) | matched(B,P)] as f32

  char* ws = (char*)d_ws;
  float* cost = (float*)ws;                                        // 4 MB
  float* iou  = (float*)(ws + (size_t)WB * WG * WP * 4);           // 4 MB
  int*   ks   = (int*)  (ws + (size_t)WB * WG * WP * 8);           // 2 KB

  dim3 g1(WP / 256, WB);
  k_cost<<<g1, 256, 0, stream>>>(preds, targets, masks, img_w, cost, iou);
  dim3 g2(WG, WB);
  k_dynk<<<g2, 256, 0, stream>>>(iou, ks);
  k_assign<<<WB, 1024, 0, stream>>>(cost, ks, masks, out);
}